// ComboLoss_78769700208763
// MI455X (gfx1250) — compile-verified
//
#include <hip/hip_runtime.h>

#define NCLS 21
#define HWPX (512 * 512)
#define NBATCH 8
#define NPIX_TOTAL (NBATCH * HWPX)

// workspace layout (floats)
#define WS_INTER 0
#define WS_PSUM  168
#define WS_TSUM  336
#define WS_CE    504
#define WS_FOCAL 505

typedef __attribute__((ext_vector_type(2))) float v2f;
typedef __attribute__((ext_vector_type(8))) float v8f;

__global__ void combo_zero_ws(float* __restrict__ ws) {
    ws[threadIdx.x] = 0.0f;   // 512 threads covers 506 used floats
}

// Main streaming kernel: grid = (GX, NBATCH), block = 256.
// Bandwidth-bound: 21 coalesced NT loads per pixel, in-register softmax,
// LDS-atomic class accumulation, one small global-atomic flush per block.
__global__ void combo_main(const float* __restrict__ pred,
                           const int* __restrict__ target,
                           float* __restrict__ ws) {
    __shared__ float inter_l[NCLS];
    __shared__ float tsum_l[NCLS];
    __shared__ float psum_l[NCLS];
    __shared__ float ce_l;
    __shared__ float focal_l;

    const int tid = threadIdx.x;
    if (tid < NCLS) { inter_l[tid] = 0.f; tsum_l[tid] = 0.f; psum_l[tid] = 0.f; }
    if (tid == NCLS) ce_l = 0.f;
    if (tid == NCLS + 1) focal_l = 0.f;
    __syncthreads();

    const int b = blockIdx.y;
    const float* __restrict__ pb = pred + (size_t)b * NCLS * HWPX;
    const int* __restrict__ tb = target + (size_t)b * HWPX;

    float psum[NCLS];
#pragma unroll
    for (int c = 0; c < NCLS; ++c) psum[c] = 0.f;
    float ce_acc = 0.f, focal_acc = 0.f;

    const int stride = gridDim.x * blockDim.x;
    for (int pix = blockIdx.x * blockDim.x + tid; pix < HWPX; pix += stride) {
        float p[NCLS];
#pragma unroll
        for (int c = 0; c < NCLS; ++c)
            p[c] = __builtin_nontemporal_load(pb + (size_t)c * HWPX + pix);

        float m = p[0];
#pragma unroll
        for (int c = 1; c < NCLS; ++c) m = fmaxf(m, p[c]);

        float s = 0.f;
#pragma unroll
        for (int c = 0; c < NCLS; ++c) {
            float e = __expf(p[c] - m);
            p[c] = e;
            s += e;
        }
        const float r = __builtin_amdgcn_rcpf(s);
        const int t = __builtin_nontemporal_load(tb + pix);

        float pt = 0.f;
#pragma unroll
        for (int c = 0; c < NCLS; ++c) {
            float pr = p[c] * r;       // prob of class c
            psum[c] += pr;
            pt = (c == t) ? pr : pt;   // v_cndmask, avoids register spill
        }
        const float ce = -__logf(pt);
        ce_acc += ce;
        const float om = 1.f - pt;
        focal_acc += om * om * ce;

        atomicAdd(&inter_l[t], pt);    // ds_add_f32
        atomicAdd(&tsum_l[t], 1.0f);
    }

#pragma unroll
    for (int c = 0; c < NCLS; ++c) atomicAdd(&psum_l[c], psum[c]);
    atomicAdd(&ce_l, ce_acc);
    atomicAdd(&focal_l, focal_acc);
    __syncthreads();

    if (tid < NCLS) {
        atomicAdd(&ws[WS_INTER + b * NCLS + tid], inter_l[tid]);
        atomicAdd(&ws[WS_PSUM + b * NCLS + tid], psum_l[tid]);
        atomicAdd(&ws[WS_TSUM + b * NCLS + tid], tsum_l[tid]);
    } else if (tid == NCLS) {
        atomicAdd(&ws[WS_CE], ce_l);
    } else if (tid == NCLS + 1) {
        atomicAdd(&ws[WS_FOCAL], focal_l);
    }
}

// Final reduction: one wave32. Dice terms accumulate into A rows 0-7,
// Jaccard terms into rows 8-15; WMMA with all-ones B turns rows into
// row-sums, and summing D's 8 VGPRs leaves the Dice total on lanes 0-15
// and Jaccard total on lanes 16-31.
__global__ void combo_final(const float* __restrict__ ws, float* __restrict__ out) {
    const int lane = threadIdx.x;            // 0..31, no divergence at WMMA
    const float S = 1.0f;
    const bool isJac = (lane & 8) != 0;      // lanes 8-15,24-31 own rows 8-15
    // unique slot pair (0..31) within this lane's group of 32 A-slots
    const int slotBase = (((lane & 7) + ((lane >> 4) & 1) * 8) << 1);

    float acc[2] = {0.f, 0.f};
#pragma unroll
    for (int pair = 0; pair < 2; ++pair) {
        for (int i = slotBase + pair; i < NBATCH * NCLS; i += 32) {
            float inter = ws[WS_INTER + i];
            float denom = ws[WS_PSUM + i] + ws[WS_TSUM + i];
            acc[pair] += isJac ? (inter + S) / (denom - inter + S)
                               : (2.f * inter + S) / (denom + S);
        }
    }

    v2f A;  A[0] = acc[0]; A[1] = acc[1];
    v2f Bv; Bv[0] = 1.0f;  Bv[1] = 1.0f;     // all-ones 4x16 B matrix
    v8f Cz = {0.f, 0.f, 0.f, 0.f, 0.f, 0.f, 0.f, 0.f};
    v8f D = __builtin_amdgcn_wmma_f32_16x16x4_f32(
        /*neg_a=*/false, A, /*neg_b=*/false, Bv,
        /*c_mod=*/(short)0, Cz, /*reuse_a=*/false, /*reuse_b=*/false);

    float rsum = 0.f;
#pragma unroll
    for (int k = 0; k < 8; ++k) rsum += D[k];   // lanes<16: dice, lanes>=16: jaccard

    float other = __shfl(rsum, 16, 32);
    if (lane == 0) {
        const float invBC = 1.0f / (float)(NBATCH * NCLS);
        float dice_loss = 1.0f - rsum * invBC;
        float jac_loss = 1.0f - other * invBC;
        const float invN = 1.0f / (float)NPIX_TOTAL;
        float ce_loss = ws[WS_CE] * invN;
        float focal_loss = ws[WS_FOCAL] * invN;
        out[0] = ce_loss + dice_loss + jac_loss + focal_loss;
    }
}

extern "C" void kernel_launch(void* const* d_in, const int* in_sizes, int n_in,
                              void* d_out, int out_size, void* d_ws, size_t ws_size,
                              hipStream_t stream) {
    const float* pred = (const float*)d_in[0];   // [8,21,512,512] f32
    const int* target = (const int*)d_in[1];     // [8,512,512] i32
    float* ws = (float*)d_ws;                    // >= 512 floats used
    float* out = (float*)d_out;                  // scalar f32

    combo_zero_ws<<<1, 512, 0, stream>>>(ws);
    combo_main<<<dim3(128, NBATCH), 256, 0, stream>>>(pred, target, ws);
    combo_final<<<1, 32, 0, stream>>>(ws, out);
}